// RewardCriterion2_3298534883602
// MI455X (gfx1250) — compile-verified
//
#include <hip/hip_runtime.h>

typedef __attribute__((ext_vector_type(2))) float v2f;
typedef __attribute__((ext_vector_type(8))) float v8f;

#define BLOCK 1024  // 32 wave32s on one WGP; whole problem in one workgroup

__global__ __launch_bounds__(BLOCK)
void reward_criterion_kernel(const float* __restrict__ logp,
                             const float* __restrict__ reward,
                             const int*   __restrict__ target,
                             float*       __restrict__ out,
                             int N, int V)
{
    __shared__ float partials[BLOCK];

    const int tid = threadIdx.x;

    // Phase 1: strided gather + multiply-accumulate. Data-dependent gather,
    // so this is plain VMEM; prefetch the streamed index/reward arrays.
    float acc = 0.0f;
    for (int i = tid; i < N; i += BLOCK) {
        const int nxt = i + BLOCK;
        if (nxt < N) {
            __builtin_prefetch(&target[nxt], 0, 0);   // global_prefetch_b8
            __builtin_prefetch(&reward[nxt], 0, 0);
        }
        const int t = target[i];
        acc += logp[(size_t)i * (size_t)V + (size_t)t] * reward[i];
    }

    partials[tid] = acc;
    __syncthreads();

    // Phase 2: wave 0 reduces all 1024 partials deterministically.
    if (tid < 32) {
        float s = 0.0f;
#pragma unroll
        for (int j = 0; j < BLOCK / 32; ++j)
            s += partials[tid + 32 * j];

#if __has_builtin(__builtin_amdgcn_wmma_f32_16x16x4_f32)
        // Reduce the 32 per-lane partials through the matrix pipe (pure f32):
        // A = ones(16x4), B VGPR0 = per-lane partial, B VGPR1 = 0, C = 0.
        // D[m,n] = sum_k B[k,n]  ->  sum_n D[0,n] == sum of all 32 partials.
        // EXEC is all-1s here: every lane of wave 0 satisfies tid < 32.
        v2f a = {1.0f, 1.0f};
        v2f b = {s, 0.0f};
        v8f c = {};
        c = __builtin_amdgcn_wmma_f32_16x16x4_f32(
                /*neg_a=*/false, a, /*neg_b=*/false, b,
                /*c_mod=*/(short)0, c, /*reuse_a=*/false, /*reuse_b=*/false);
        // c[0]: lanes 0-15 hold D[0, lane]; lanes 16-31 duplicate (row M=8).
        float tsum = c[0];
        tsum += __shfl_xor(tsum, 1);
        tsum += __shfl_xor(tsum, 2);
        tsum += __shfl_xor(tsum, 4);
        tsum += __shfl_xor(tsum, 8);   // lane 0: sum over N=0..15 -> total
#else
        // Fallback: plain wave32 xor-shuffle tree.
        float tsum = s;
        tsum += __shfl_xor(tsum, 1);
        tsum += __shfl_xor(tsum, 2);
        tsum += __shfl_xor(tsum, 4);
        tsum += __shfl_xor(tsum, 8);
        tsum += __shfl_xor(tsum, 16);
#endif
        if (tid == 0) out[0] = -tsum;
    }
}

extern "C" void kernel_launch(void* const* d_in, const int* in_sizes, int n_in,
                              void* d_out, int out_size, void* d_ws, size_t ws_size,
                              hipStream_t stream) {
    // setup_inputs order: seqLogprobs (f32), reward (f32), batchsize_cap (int scalar), target (int)
    const float* logp   = (const float*)d_in[0];
    const float* reward = (const float*)d_in[1];
    const int*   target = (const int*)d_in[3];

    const int N = in_sizes[1];            // 32 * 179 = 5728
    const int V = in_sizes[0] / N;        // 9487

    reward_criterion_kernel<<<1, BLOCK, 0, stream>>>(
        logp, reward, target, (float*)d_out, N, V);
}